// attention_pooling_layer_68384469287307
// MI455X (gfx1250) — compile-verified
//
#include <hip/hip_runtime.h>
#include <math.h>

// Problem constants (match reference)
#define B_    32
#define SV_   4096
#define SQ_   1152
#define D_    128
#define FD_   96
#define POOL_ 12            // SQ/FD
#define NCH_  16            // chunks over SV for parallelism
#define CHS_  (SV_ / NCH_)  // 256
#define D4_   (D_ / 4)      // 32 float4 per row

// Workspace layout (float offsets; all multiples of 32 -> float4-aligned)
#define OFF_QS     ((size_t)0)                 // B*D
#define OFF_SUMQ   (OFF_QS + B_*D_)            // B
#define OFF_HM     (OFF_SUMQ + B_)             // 3 (+pad)
#define OFF_S1P    (OFF_HM + 32)               // B*NCH*D
#define OFF_S2P    (OFF_S1P + B_*NCH_*D_)
#define OFF_MNP    (OFF_S2P + B_*NCH_*D_)
#define OFF_MXP    (OFF_MNP + B_*NCH_*D_)
#define OFF_VS     (OFF_MXP + B_*NCH_*D_)      // B*D  (S1 = vs)
#define OFF_ALPHA  (OFF_VS + B_*D_)            // B*3*D
#define OFF_MV     (OFF_ALPHA + B_*3*D_)       // B*3*D
#define OFF_Z      (OFF_MV + B_*3*D_)          // B*3*D
#define OFF_VSUM   (OFF_Z + B_*3*D_)           // B
#define OFF_ATT0   (OFF_VSUM + B_)             // B*D
#define OFF_AROW   (OFF_ATT0 + B_*D_)          // B
#define OFF_A2ROW  (OFF_AROW + B_)             // B
#define OFF_LI     (OFF_A2ROW + B_)            // B
#define OFF_POOL   (OFF_LI + B_)               // B*FD
#define OFF_MEAN   (OFF_POOL + B_*FD_)         // FD
#define OFF_INV    (OFF_MEAN + FD_)            // FD

typedef __attribute__((ext_vector_type(2))) float v2f;
typedef __attribute__((ext_vector_type(4))) float v4f;
typedef __attribute__((ext_vector_type(8))) float v8f;

__device__ __forceinline__ float wave_reduce_add(float x) {
  // wave32 butterfly reduction
  #pragma unroll
  for (int off = 16; off > 0; off >>= 1) x += __shfl_xor(x, off, 32);
  return x;
}

__device__ __forceinline__ v4f v4_min(v4f a, v4f b) {
  v4f r; r.x = fminf(a.x,b.x); r.y = fminf(a.y,b.y); r.z = fminf(a.z,b.z); r.w = fminf(a.w,b.w);
  return r;
}
__device__ __forceinline__ v4f v4_max(v4f a, v4f b) {
  v4f r; r.x = fmaxf(a.x,b.x); r.y = fmaxf(a.y,b.y); r.z = fmaxf(a.z,b.z); r.w = fmaxf(a.w,b.w);
  return r;
}

// K1: qs[b,k] = sum of q.flat[b, k*SQ : (k+1)*SQ]; sumq[b] = sum_k qs; hm[c] = sum h_mat
// b128 loads: SQ = 1152 floats = 288 float4 = 9 iterations per lane.
__global__ void k1_qstats(const float* __restrict__ q,
                          const float* __restrict__ h_mat,
                          float* __restrict__ ws) {
  int b = blockIdx.x;
  int tid = threadIdx.x;            // 256 threads = 8 waves
  int wave = tid >> 5, lane = tid & 31;
  __shared__ float lqs[D_];
  const float* qb = q + (size_t)b * SQ_ * D_;
  #pragma unroll 1
  for (int kk = 0; kk < 16; ++kk) {
    int k = wave * 16 + kk;
    const v4f* row = (const v4f*)(qb + (size_t)k * SQ_);
    float s = 0.f;
    #pragma unroll
    for (int j = 0; j < SQ_ / 4 / 32; ++j) {   // 9
      v4f x = row[lane + 32 * j];
      s += (x.x + x.y) + (x.z + x.w);
    }
    s = wave_reduce_add(s);
    if (lane == 0) { ws[OFF_QS + (size_t)b * D_ + k] = s; lqs[k] = s; }
  }
  __syncthreads();
  if (wave == 0) {
    float s = lqs[lane] + lqs[lane + 32] + lqs[lane + 64] + lqs[lane + 96];
    s = wave_reduce_add(s);
    if (lane == 0) ws[OFF_SUMQ + b] = s;
  }
  if (b == 0 && tid < 3) {
    float s = 0.f;
    for (int x = 0; x < FD_; ++x) s += h_mat[tid * FD_ + x];
    ws[OFF_HM + tid] = s;
  }
}

// K2a: partial S1/S2/min/max of v over an SV chunk, per (b,d).
// Each lane owns 4 consecutive d (b128 loads); 8 s-slices per block.
__global__ void k2a_vpartial(const float* __restrict__ v, float* __restrict__ ws) {
  int blk = blockIdx.x;             // b*NCH + ch
  int b = blk / NCH_, ch = blk % NCH_;
  int tid = threadIdx.x;            // 256
  int lane = tid & 31;              // d-group (4 floats)
  int sp = tid >> 5;                // 0..7 s-slice
  const v4f* vb = (const v4f*)(v + (size_t)b * SV_ * D_);
  v4f s1 = {0.f,0.f,0.f,0.f}, s2 = {0.f,0.f,0.f,0.f};
  v4f mn = { 3.0e38f, 3.0e38f, 3.0e38f, 3.0e38f};
  v4f mx = {-3.0e38f,-3.0e38f,-3.0e38f,-3.0e38f};
  #pragma unroll 4
  for (int s = ch * CHS_ + sp; s < (ch + 1) * CHS_; s += 8) {
    v4f x = vb[(size_t)s * D4_ + lane];
    s1 += x;
    s2 += x * x;
    mn = v4_min(mn, x); mx = v4_max(mx, x);
  }
  __shared__ v4f r1[8][32], r2[8][32], rmn[8][32], rmx[8][32];
  r1[sp][lane] = s1; r2[sp][lane] = s2; rmn[sp][lane] = mn; rmx[sp][lane] = mx;
  __syncthreads();
  if (tid < 32) {
    v4f a1 = r1[0][tid], a2 = r2[0][tid], an = rmn[0][tid], ax = rmx[0][tid];
    #pragma unroll
    for (int j = 1; j < 8; ++j) {
      a1 += r1[j][tid]; a2 += r2[j][tid];
      an = v4_min(an, rmn[j][tid]); ax = v4_max(ax, rmx[j][tid]);
    }
    size_t o = (size_t)blk * D4_ + tid;
    ((v4f*)(ws + OFF_S1P))[o] = a1;
    ((v4f*)(ws + OFF_S2P))[o] = a2;
    ((v4f*)(ws + OFF_MNP))[o] = an;
    ((v4f*)(ws + OFF_MXP))[o] = ax;
  }
}

// K2b: combine partials -> S1,S2,min,max; alpha = K/norm; m = max_s(alpha*v); zero Z; vsum
__global__ void k2b_alpha(const float* __restrict__ h_bias, float* __restrict__ ws) {
  int b = blockIdx.x, d = threadIdx.x;   // 128 threads
  float s1 = 0.f, s2 = 0.f, mn = 3.0e38f, mx = -3.0e38f;
  for (int ch = 0; ch < NCH_; ++ch) {
    size_t o = (size_t)(b * NCH_ + ch) * D_ + d;
    s1 += ws[OFF_S1P + o]; s2 += ws[OFF_S2P + o];
    mn = fminf(mn, ws[OFF_MNP + o]); mx = fmaxf(mx, ws[OFF_MXP + o]);
  }
  ws[OFF_VS + (size_t)b * D_ + d] = s1;
  float sumq = ws[OFF_SUMQ + b];
  #pragma unroll
  for (int c = 0; c < 3; ++c) {
    float K  = ws[OFF_HM + c] * sumq;
    float hb = h_bias[c];
    float n2 = K * K * s2 + 2.f * K * hb * s1 + (float)SV_ * hb * hb;
    float alpha = K * rsqrtf(n2);                        // K / ||att row||
    float m = (alpha >= 0.f) ? alpha * mx : alpha * mn;  // max_s alpha*v (stability)
    size_t o = ((size_t)b * 3 + c) * D_ + d;
    ws[OFF_ALPHA + o] = alpha; ws[OFF_MV + o] = m; ws[OFF_Z + o] = 0.f;
  }
  __shared__ float row[D_];
  row[d] = s1;
  __syncthreads();
  if (d < 32) {
    float s = row[d] + row[d + 32] + row[d + 64] + row[d + 96];
    s = wave_reduce_add(s);
    if (d == 0) ws[OFF_VSUM + b] = s;
  }
}

// K2c: Z[b,c,d] += sum_s exp(alpha*v - m) over an SV chunk.
// b128 loads of v; 12 v_exp_f32 per float4 (3 channels x 4 d-columns).
__global__ void k2c_zpass(const float* __restrict__ v, float* __restrict__ ws) {
  int blk = blockIdx.x;
  int b = blk / NCH_, ch = blk % NCH_;
  int tid = threadIdx.x;
  int lane = tid & 31;              // d-group (4 floats)
  int sp = tid >> 5;                // 0..7
  const v4f* vb = (const v4f*)(v + (size_t)b * SV_ * D_);
  size_t ob4 = (size_t)b * 3 * D4_ + lane;   // float4 index into [B,3,D]
  const v4f* wa = (const v4f*)(ws + OFF_ALPHA);
  const v4f* wm = (const v4f*)(ws + OFF_MV);
  v4f a0 = wa[ob4],           m0 = wm[ob4];
  v4f a1 = wa[ob4 + D4_],     m1 = wm[ob4 + D4_];
  v4f a2 = wa[ob4 + 2*D4_],   m2 = wm[ob4 + 2*D4_];
  v4f z0 = {0.f,0.f,0.f,0.f}, z1 = z0, z2 = z0;
  #pragma unroll 2
  for (int s = ch * CHS_ + sp; s < (ch + 1) * CHS_; s += 8) {
    v4f x = vb[(size_t)s * D4_ + lane];
    z0.x += __expf(fmaf(a0.x, x.x, -m0.x));
    z0.y += __expf(fmaf(a0.y, x.y, -m0.y));
    z0.z += __expf(fmaf(a0.z, x.z, -m0.z));
    z0.w += __expf(fmaf(a0.w, x.w, -m0.w));
    z1.x += __expf(fmaf(a1.x, x.x, -m1.x));
    z1.y += __expf(fmaf(a1.y, x.y, -m1.y));
    z1.z += __expf(fmaf(a1.z, x.z, -m1.z));
    z1.w += __expf(fmaf(a1.w, x.w, -m1.w));
    z2.x += __expf(fmaf(a2.x, x.x, -m2.x));
    z2.y += __expf(fmaf(a2.y, x.y, -m2.y));
    z2.z += __expf(fmaf(a2.z, x.z, -m2.z));
    z2.w += __expf(fmaf(a2.w, x.w, -m2.w));
  }
  __shared__ v4f r0[8][32], r1[8][32], r2[8][32];
  r0[sp][lane] = z0; r1[sp][lane] = z1; r2[sp][lane] = z2;
  __syncthreads();
  if (tid < 32) {
    v4f c0 = r0[0][tid], c1 = r1[0][tid], c2 = r2[0][tid];
    #pragma unroll
    for (int j = 1; j < 8; ++j) { c0 += r0[j][tid]; c1 += r1[j][tid]; c2 += r2[j][tid]; }
    size_t ob = (size_t)b * 3 * D_ + tid * 4;
    atomicAdd(&ws[OFF_Z + ob + 0], c0.x);
    atomicAdd(&ws[OFF_Z + ob + 1], c0.y);
    atomicAdd(&ws[OFF_Z + ob + 2], c0.z);
    atomicAdd(&ws[OFF_Z + ob + 3], c0.w);
    atomicAdd(&ws[OFF_Z + ob + D_ + 0], c1.x);
    atomicAdd(&ws[OFF_Z + ob + D_ + 1], c1.y);
    atomicAdd(&ws[OFF_Z + ob + D_ + 2], c1.z);
    atomicAdd(&ws[OFF_Z + ob + D_ + 3], c1.w);
    atomicAdd(&ws[OFF_Z + ob + 2*D_ + 0], c2.x);
    atomicAdd(&ws[OFF_Z + ob + 2*D_ + 1], c2.y);
    atomicAdd(&ws[OFF_Z + ob + 2*D_ + 2], c2.z);
    atomicAdd(&ws[OFF_Z + ob + 2*D_ + 3], c2.w);
  }
}

// K2d: att rows s=0,1,2 summed over c; per-batch aggregates Arow, A2row, li
__global__ void k2d_att(const float* __restrict__ v, float* __restrict__ ws) {
  int b = blockIdx.x, d = threadIdx.x;   // 128 threads
  size_t ob = (size_t)b * 3 * D_ + d;
  float a[3], m[3], Z[3];
  #pragma unroll
  for (int c = 0; c < 3; ++c) {
    a[c] = ws[OFF_ALPHA + ob + (size_t)c * D_];
    m[c] = ws[OFF_MV    + ob + (size_t)c * D_];
    Z[c] = ws[OFF_Z     + ob + (size_t)c * D_];
  }
  const float* vb = v + (size_t)b * SV_ * D_;
  float att[3];
  #pragma unroll
  for (int i = 0; i < 3; ++i) {
    float x = vb[(size_t)i * D_ + d];
    float t = 0.f;
    #pragma unroll
    for (int c = 0; c < 3; ++c) t += __expf(fmaf(a[c], x, -m[c])) / Z[c];
    att[i] = t;
  }
  ws[OFF_ATT0 + (size_t)b * D_ + d] = att[0];
  float qs = ws[OFF_QS + (size_t)b * D_ + d];
  __shared__ float ra[D_], rb[D_], rc[D_];
  ra[d] = att[0]; rb[d] = att[0] * att[0]; rc[d] = (att[1] + att[2]) * qs;
  __syncthreads();
  if (d < 32) {
    float x = ra[d] + ra[d + 32] + ra[d + 64] + ra[d + 96]; x = wave_reduce_add(x);
    float y = rb[d] + rb[d + 32] + rb[d + 64] + rb[d + 96]; y = wave_reduce_add(y);
    float z = rc[d] + rc[d + 32] + rc[d + 64] + rc[d + 96]; z = wave_reduce_add(z);
    if (d == 0) {
      ws[OFF_AROW  + b] = x;
      ws[OFF_A2ROW + b] = y;
      ws[OFF_LI    + b] = ws[OFF_VSUM + b] * z;
    }
  }
}

// K3: pool_inner[b,f] = sum_k vs[b,k] * Qpool[b,k,f] via V_WMMA_F32_16X16X4_F32.
// A (16x4) = vs chunk broadcast across M rows; B (4x16) = pooled-q tile; take row M=0.
// One wave per (b, f-tile); EXEC all ones, no divergence around the WMMA loop.
__global__ void k3_pool_wmma(const float* __restrict__ q, float* __restrict__ ws) {
  int blk = blockIdx.x;                  // b*6 + ftile
  int b = blk / 6, ft = blk % 6;
  int lane = threadIdx.x;                // 32
  int half = lane >> 4, col = lane & 15;
  const float* qb = q + (size_t)b * SQ_ * D_;
  const float* vs = ws + OFF_VS + (size_t)b * D_;
  const float inv12 = 1.0f / (float)POOL_;
  int f = ft * 16 + col;
  v8f acc = {};
  #pragma unroll 1
  for (int kk = 0; kk < 32; ++kk) {
    // ISA 16x4 f32 A layout: VGPR0 = K0(lanes0-15)/K2(lanes16-31), VGPR1 = K1/K3
    int kA0 = kk * 4 + (half ? 2 : 0);
    int kA1 = kk * 4 + (half ? 3 : 1);
    v2f A, Bm;
    A.x = vs[kA0];
    A.y = vs[kA1];
    // B 4x16: row K striped across lanes; Qpool[k,f] = mean of 12 contiguous q floats
    const float* p0 = qb + (size_t)kA0 * SQ_ + f * POOL_;
    const float* p1 = qb + (size_t)kA1 * SQ_ + f * POOL_;
    float s0 = 0.f, s1 = 0.f;
    #pragma unroll
    for (int j = 0; j < POOL_; ++j) { s0 += p0[j]; s1 += p1[j]; }
    Bm.x = s0 * inv12;
    Bm.y = s1 * inv12;
    acc = __builtin_amdgcn_wmma_f32_16x16x4_f32(
        /*neg_a=*/false, A, /*neg_b=*/false, Bm,
        /*c_mod=*/(short)0, acc, /*reuse_a=*/false, /*reuse_b=*/false);
  }
  // D row M=0 lives in VGPR0 lanes 0..15
  if (lane < 16) ws[OFF_POOL + (size_t)b * FD_ + ft * 16 + lane] = acc[0];
}

// K4a: closed-form BatchNorm stats per channel f from per-batch aggregates
__global__ void k4a_bnstats(float* __restrict__ ws) {
  int f = threadIdx.x;   // 128 launched, 96 active
  if (f >= FD_) return;
  float s = 0.f, s2 = 0.f;
  for (int b = 0; b < B_; ++b) {
    float p   = ws[OFF_POOL + (size_t)b * FD_ + f];
    float L   = ws[OFF_LI + b];
    float Ar  = ws[OFF_AROW + b];
    float A2r = ws[OFF_A2ROW + b];
    s  += p * Ar + (float)D_ * L;
    s2 += p * p * A2r + 2.f * p * L * Ar + (float)D_ * L * L;
  }
  float n = (float)(B_ * D_);
  float mean = s / n;
  float var  = s2 / n - mean * mean;
  ws[OFF_MEAN + f] = mean;
  ws[OFF_INV + f]  = rsqrtf(var + 1e-5f);
}

// K4b: out[b,f,d] = ((att0[b,d]*pool[b,f] + li[b]) - mean_f) * inv_f * gamma_f + beta_f
// Vectorized: one float4 (4 consecutive d) per thread; b128 stores.
__global__ void k4b_out(const float* __restrict__ gamma, const float* __restrict__ beta,
                        const float* __restrict__ ws, float* __restrict__ out) {
  int idx = blockIdx.x * blockDim.x + threadIdx.x;   // over B*FD*D/4
  if (idx >= B_ * FD_ * D4_) return;
  int l = idx & (D4_ - 1);
  int f = (idx / D4_) % FD_;
  int b = idx / (D4_ * FD_);
  v4f a0 = ((const v4f*)(ws + OFF_ATT0 + (size_t)b * D_))[l];
  float p = ws[OFF_POOL + (size_t)b * FD_ + f];
  float L = ws[OFF_LI + b];
  float mu = ws[OFF_MEAN + f];
  float sc = ws[OFF_INV + f] * gamma[f];
  float bt = beta[f];
  v4f r;
  r.x = (fmaf(a0.x, p, L) - mu) * sc + bt;
  r.y = (fmaf(a0.y, p, L) - mu) * sc + bt;
  r.z = (fmaf(a0.z, p, L) - mu) * sc + bt;
  r.w = (fmaf(a0.w, p, L) - mu) * sc + bt;
  ((v4f*)out)[idx] = r;
}

extern "C" void kernel_launch(void* const* d_in, const int* in_sizes, int n_in,
                              void* d_out, int out_size, void* d_ws, size_t ws_size,
                              hipStream_t stream) {
  (void)in_sizes; (void)n_in; (void)out_size; (void)ws_size;
  const float* q      = (const float*)d_in[0];
  const float* v      = (const float*)d_in[1];
  const float* h_mat  = (const float*)d_in[2];
  const float* h_bias = (const float*)d_in[3];
  const float* gamma  = (const float*)d_in[4];
  const float* beta   = (const float*)d_in[5];
  float* out = (float*)d_out;
  float* ws  = (float*)d_ws;

  k1_qstats   <<<B_,        256, 0, stream>>>(q, h_mat, ws);
  k2a_vpartial<<<B_ * NCH_, 256, 0, stream>>>(v, ws);
  k2b_alpha   <<<B_,        D_,  0, stream>>>(h_bias, ws);
  k2c_zpass   <<<B_ * NCH_, 256, 0, stream>>>(v, ws);
  k2d_att     <<<B_,        D_,  0, stream>>>(v, ws);
  k3_pool_wmma<<<B_ * 6,    32,  0, stream>>>(q, ws);
  k4a_bnstats <<<1,         128, 0, stream>>>(ws);
  int total4 = B_ * FD_ * D4_;
  k4b_out     <<<(total4 + 255) / 256, 256, 0, stream>>>(gamma, beta, ws, out);
}